// LinearSystem_85598698209590
// MI455X (gfx1250) — compile-verified
//
#include <hip/hip_runtime.h>

// LinearSystem scan on MI455X (gfx1250, wave32, WMMA).
// f32 path via V_WMMA_F32_16X16X4_F32; chunked-scan decomposition for parallelism.

#define S_DIM   256
#define I_DIM   64
#define O_DIM   64
#define BATCH   64
#define TLEN    2048
#define CHUNK   64
#define NCHUNK  (TLEN / CHUNK)   // 32
#define MATSZ   (S_DIM * S_DIM)  // 65536 floats per 256x256 matrix

#define LDP  260   // padded LDS row stride for 256-wide tiles (kills bank conflicts)
#define LDPI 68    // padded LDS row stride for 64-wide tiles

typedef __attribute__((ext_vector_type(2))) float v2f;
typedef __attribute__((ext_vector_type(4))) float v4f;
typedef __attribute__((ext_vector_type(8))) float v8f;

// ---- WMMA f32 16x16x4 wrapper --------------------------------------------
__device__ __forceinline__ v8f wmma4(v2f a, v2f b, v8f c) {
  // (neg_a, A, neg_b, B, c_mod, C, reuse_a, reuse_b)
  return __builtin_amdgcn_wmma_f32_16x16x4_f32(false, a, false, b, (short)0, c,
                                               false, false);
}

// A-fragment 16x4 (MxK) from row-major [m][k]: lanes 0-15 K={0,1}, 16-31 K={2,3}
__device__ __forceinline__ v2f frag_a(const float* base, int ld, int k0, int lane) {
  int m = lane & 15, kh = lane >> 4;
  const float* p = base + m * ld + k0 + 2 * kh;
  v2f r; r.x = p[0]; r.y = p[1]; return r;
}
// B-fragment for D = A * W^T, W row-major [n][k] (W-row strip starts at base)
__device__ __forceinline__ v2f frag_bT(const float* W, int ld, int k0, int lane) {
  int n = lane & 15, kh = lane >> 4;
  const float* p = W + n * ld + k0 + 2 * kh;
  v2f r; r.x = p[0]; r.y = p[1]; return r;
}
// B-fragment for plain D = A * B, B row-major [k][n] (col strip starts at base)
__device__ __forceinline__ v2f frag_b(const float* B, int ld, int k0, int lane) {
  int n = lane & 15, kh = lane >> 4;
  v2f r;
  r.x = B[(k0 + 2 * kh) * ld + n];
  r.y = B[(k0 + 2 * kh + 1) * ld + n];
  return r;
}
// C/D 16x16: VGPR v, lanes 0-15 -> M=v, lanes 16-31 -> M=v+8
__device__ __forceinline__ void store_tile(float* dst, int ld, v8f c, int lane) {
  int n = lane & 15, half = lane >> 4;
#pragma unroll
  for (int v = 0; v < 8; ++v) dst[(v + 8 * half) * ld + n] = c[v];
}

__device__ __forceinline__ v8f zero8() {
  v8f z = {0.f, 0.f, 0.f, 0.f, 0.f, 0.f, 0.f, 0.f};
  return z;
}

// ---- P_1 = F --------------------------------------------------------------
__global__ void copy_f_kernel(const float* __restrict__ F, float* __restrict__ P) {
  int i = blockIdx.x * blockDim.x + threadIdx.x;  // 16384 float4s
  ((v4f*)P)[i] = ((const v4f*)F)[i];
}

// ---- power doubling: P_{s+g+1} = P_s * P_{g+1}, g = blockIdx.x -----------
__global__ __launch_bounds__(512, 1) void powers_kernel(float* __restrict__ Pbase,
                                                        int s) {
  extern __shared__ float smem[];
  float* sA = smem;  // [256][LDP]
  const int g = blockIdx.x;
  const float* A = Pbase + (size_t)(s - 1) * MATSZ;
  const float* B = Pbase + (size_t)g * MATSZ;
  float* D = Pbase + (size_t)(s + g) * MATSZ;
  const int tid = threadIdx.x, lane = tid & 31, w = tid >> 5;

  for (int i = tid; i < 16384; i += 512) {  // stage A (256x256) padded
    int r = i >> 6, c4 = i & 63;
    ((v4f*)(sA + r * LDP))[c4] = ((const v4f*)(A + r * 256))[c4];
  }
  v2f bres[64];  // resident B column-strip [*, 16w..16w+16)
#pragma unroll
  for (int kt = 0; kt < 64; ++kt) bres[kt] = frag_b(B + w * 16, 256, 4 * kt, lane);
  __syncthreads();

#pragma unroll 1
  for (int mt = 0; mt < 16; ++mt) {
    v8f acc = zero8();
#pragma unroll
    for (int kt = 0; kt < 64; ++kt)
      acc = wmma4(frag_a(sA + mt * 16 * LDP, LDP, 4 * kt, lane), bres[kt], acc);
    store_tile(D + mt * 16 * 256 + w * 16, 256, acc, lane);
  }
}

// ---- Phase A: per-chunk local scan with fused drive GEMMs ----------------
// block = 512 threads (16 waves); wave w owns state dims [16w, 16w+16)
__global__ __launch_bounds__(512, 1) void scan_kernel(
    const float* __restrict__ state0, const float* __restrict__ inputs,
    const float* __restrict__ epsW, const float* __restrict__ F,
    const float* __restrict__ Bmat, const float* __restrict__ sqrtSW,
    float* __restrict__ states) {
  extern __shared__ float smem[];
  float* sSW  = smem;                 // [256][LDP]  sqrt_S_W
  float* sCur0 = sSW + 256 * LDP;     // [16][LDP]   state double buffer
  float* sCur1 = sCur0 + 16 * LDP;    // [16][LDP]
  float* sEps = sCur1 + 16 * LDP;     // [16][LDP]   eps_W tile
  float* sInp = sEps + 16 * LDP;      // [16][LDPI]  inputs tile
  const int tid = threadIdx.x, lane = tid & 31, w = tid >> 5;
  const int chunk = blockIdx.x >> 2, btile = blockIdx.x & 3;
  const int b0 = btile * 16, t0 = chunk * CHUNK;

  for (int i = tid; i < 16384; i += 512) {  // stage sqrt_S_W once
    int r = i >> 6, c4 = i & 63;
    ((v4f*)(sSW + r * LDP))[c4] = ((const v4f*)(sqrtSW + r * 256))[c4];
  }
  if (chunk == 0) {  // true initial state; other chunks start at zero
    for (int i = tid; i < 1024; i += 512) {
      int m = i >> 6, c4 = i & 63;
      ((v4f*)(sCur0 + m * LDP))[c4] = ((const v4f*)(state0 + (b0 + m) * 256))[c4];
    }
  } else {
    for (int i = tid; i < 16 * LDP; i += 512) sCur0[i] = 0.0f;
  }

  v2f fF[64], fB[16];  // resident row strips of F and B_mat for this wave
#pragma unroll
  for (int kt = 0; kt < 64; ++kt) fF[kt] = frag_bT(F + w * 16 * 256, 256, 4 * kt, lane);
#pragma unroll
  for (int kt = 0; kt < 16; ++kt) fB[kt] = frag_bT(Bmat + w * 16 * 64, 64, 4 * kt, lane);

  float* cur = sCur0;
  float* nxt = sCur1;
#pragma unroll 1
  for (int k = 0; k < CHUNK; ++k) {
    const int t = t0 + k;
    for (int i = tid; i < 1024; i += 512) {  // stage eps_W[16][256]
      int m = i >> 6, c4 = i & 63;
      ((v4f*)(sEps + m * LDP))[c4] =
          ((const v4f*)(epsW + ((size_t)(b0 + m) * TLEN + t) * 256))[c4];
    }
    if (tid < 256) {  // stage inputs[16][64]
      int m = tid >> 4, c4 = tid & 15;
      ((v4f*)(sInp + m * LDPI))[c4] =
          ((const v4f*)(inputs + ((size_t)(b0 + m) * TLEN + t) * 64))[c4];
    }
    if (k + 1 < CHUNK && tid < 16)  // global_prefetch_b8 next eps_W rows
      __builtin_prefetch(epsW + ((size_t)(b0 + tid) * TLEN + (t + 1)) * 256, 0, 0);
    __syncthreads();

    v8f acc = zero8();
#pragma unroll
    for (int kt = 0; kt < 16; ++kt)  // inputs @ Bmat^T   (K=64)
      acc = wmma4(frag_a(sInp, LDPI, 4 * kt, lane), fB[kt], acc);
#pragma unroll
    for (int kt = 0; kt < 64; ++kt)  // eps_W @ sqrtSW^T  (K=256)
      acc = wmma4(frag_a(sEps, LDP, 4 * kt, lane),
                  frag_bT(sSW + w * 16 * LDP, LDP, 4 * kt, lane), acc);
#pragma unroll
    for (int kt = 0; kt < 64; ++kt)  // s_cur @ F^T       (K=256)
      acc = wmma4(frag_a(cur, LDP, 4 * kt, lane), fF[kt], acc);

    store_tile(nxt + w * 16, LDP, acc, lane);
    store_tile(states + (size_t)b0 * TLEN * 256 + (size_t)t * 256 + w * 16,
               TLEN * 256, acc, lane);
    __syncthreads();
    float* tmp = cur; cur = nxt; nxt = tmp;
  }
}

// ---- Phase B: chunk-entry carries, s_in(c+1) = F^C s_in(c) + lend(c) -----
__global__ __launch_bounds__(512, 1) void carry_kernel(
    const float* __restrict__ states, const float* __restrict__ PC,
    float* __restrict__ carry /* [NCHUNK-1][BATCH][256] */) {
  extern __shared__ float smem[];
  float* sPrev = smem;  // [64][LDP]
  const int tid = threadIdx.x, lane = tid & 31, w = tid >> 5;

  for (int i = tid; i < 4096; i += 512) {  // s_in(1) = exact states[:, C-1, :]
    int b = i >> 6, c4 = i & 63;
    v4f val = ((const v4f*)(states + ((size_t)b * TLEN + (CHUNK - 1)) * 256))[c4];
    ((v4f*)(sPrev + b * LDP))[c4] = val;
    ((v4f*)(carry + b * 256))[c4] = val;
  }
  v2f fPC[64];  // resident F^C row strip
#pragma unroll
  for (int kt = 0; kt < 64; ++kt) fPC[kt] = frag_bT(PC + w * 16 * 256, 256, 4 * kt, lane);
  __syncthreads();

#pragma unroll 1
  for (int c = 2; c < NCHUNK; ++c) {
    v8f acc[4];
#pragma unroll
    for (int mt = 0; mt < 4; ++mt) {
      v8f a = zero8();
#pragma unroll
      for (int kt = 0; kt < 64; ++kt)
        a = wmma4(frag_a(sPrev + mt * 16 * LDP, LDP, 4 * kt, lane), fPC[kt], a);
      int n = lane & 15, half = lane >> 4;
#pragma unroll
      for (int v = 0; v < 8; ++v) {  // + local end-state of chunk c-1
        int b = mt * 16 + v + 8 * half;
        a[v] += states[((size_t)b * TLEN + (c * CHUNK - 1)) * 256 + w * 16 + n];
      }
      acc[mt] = a;
    }
    __syncthreads();
#pragma unroll
    for (int mt = 0; mt < 4; ++mt) {
      store_tile(sPrev + mt * 16 * LDP + w * 16, LDP, acc[mt], lane);
      store_tile(carry + (size_t)(c - 1) * BATCH * 256 + mt * 16 * 256 + w * 16,
                 256, acc[mt], lane);
    }
    __syncthreads();
  }
}

// ---- Phase C: states[cC+k] += s_in(c) @ (F^{k+1})^T ----------------------
__global__ __launch_bounds__(512, 1) void fixup_kernel(
    float* __restrict__ states, const float* __restrict__ Pbase,
    const float* __restrict__ carry) {
  extern __shared__ float smem[];
  float* sP = smem;             // [256][LDP]
  float* sIn = sP + 256 * LDP;  // [16][LDP]
  const int tid = threadIdx.x, lane = tid & 31, w = tid >> 5;
  const int c = (blockIdx.x >> 2) + 1, btile = blockIdx.x & 3;
  const int b0 = btile * 16;

  for (int i = tid; i < 1024; i += 512) {  // stage s_in(c) tile
    int m = i >> 6, c4 = i & 63;
    ((v4f*)(sIn + m * LDP))[c4] =
        ((const v4f*)(carry + ((size_t)(c - 1) * BATCH + b0 + m) * 256))[c4];
  }
#pragma unroll 1
  for (int k = 0; k < CHUNK; ++k) {
    const float* Pk = Pbase + (size_t)k * MATSZ;  // F^{k+1}, L2-resident
    for (int i = tid; i < 16384; i += 512) {
      int r = i >> 6, c4 = i & 63;
      ((v4f*)(sP + r * LDP))[c4] = ((const v4f*)(Pk + r * 256))[c4];
    }
    __syncthreads();
    v8f acc = zero8();
#pragma unroll
    for (int kt = 0; kt < 64; ++kt)
      acc = wmma4(frag_a(sIn, LDP, 4 * kt, lane),
                  frag_bT(sP + w * 16 * LDP, LDP, 4 * kt, lane), acc);
    const int t = c * CHUNK + k;
    int n = lane & 15, half = lane >> 4;
    float* dst = states + (size_t)b0 * TLEN * 256 + (size_t)t * 256 + w * 16 + n;
#pragma unroll
    for (int v = 0; v < 8; ++v) dst[(size_t)(v + 8 * half) * TLEN * 256] += acc[v];
    __syncthreads();
  }
}

// ---- Observations: states @ H^T + eps_V @ sqrtSV^T -----------------------
__global__ __launch_bounds__(512, 1) void obs_kernel(
    const float* __restrict__ states, const float* __restrict__ epsV,
    const float* __restrict__ H, const float* __restrict__ sqrtSV,
    float* __restrict__ obs) {
  extern __shared__ float smem[];
  float* sH  = smem;              // [64][LDP]
  float* sSV = sH + 64 * LDP;     // [64][LDPI]
  float* sS  = sSV + 64 * LDPI;   // [4][16][LDP]
  float* sEV = sS + 4 * 16 * LDP; // [4][16][LDPI]
  const int tid = threadIdx.x, lane = tid & 31, w = tid >> 5;
  const int btile = blockIdx.x & 3, tg = blockIdx.x >> 2;
  const int b0 = btile * 16, tBase = tg * 4;

  for (int i = tid; i < 4096; i += 512) {  // H
    int r = i >> 6, c4 = i & 63;
    ((v4f*)(sH + r * LDP))[c4] = ((const v4f*)(H + r * 256))[c4];
  }
  for (int i = tid; i < 1024; i += 512) {  // sqrt_S_V
    int r = i >> 4, c4 = i & 15;
    ((v4f*)(sSV + r * LDPI))[c4] = ((const v4f*)(sqrtSV + r * 64))[c4];
  }
  for (int i = tid; i < 4096; i += 512) {  // states [4][16][256]
    int tt = i >> 10, rem = i & 1023, m = rem >> 6, c4 = rem & 63;
    ((v4f*)(sS + (tt * 16 + m) * LDP))[c4] =
        ((const v4f*)(states + ((size_t)(b0 + m) * TLEN + tBase + tt) * 256))[c4];
  }
  for (int i = tid; i < 1024; i += 512) {  // eps_V [4][16][64]
    int tt = i >> 8, rem = i & 255, m = rem >> 4, c4 = rem & 15;
    ((v4f*)(sEV + (tt * 16 + m) * LDPI))[c4] =
        ((const v4f*)(epsV + ((size_t)(b0 + m) * TLEN + tBase + tt) * 64))[c4];
  }
  __syncthreads();

  const int tt = w >> 2, nt = w & 3;
  v8f acc = zero8();
#pragma unroll
  for (int kt = 0; kt < 64; ++kt)
    acc = wmma4(frag_a(sS + tt * 16 * LDP, LDP, 4 * kt, lane),
                frag_bT(sH + nt * 16 * LDP, LDP, 4 * kt, lane), acc);
#pragma unroll
  for (int kt = 0; kt < 16; ++kt)
    acc = wmma4(frag_a(sEV + tt * 16 * LDPI, LDPI, 4 * kt, lane),
                frag_bT(sSV + nt * 16 * LDPI, LDPI, 4 * kt, lane), acc);
  store_tile(obs + (size_t)b0 * TLEN * 64 + (size_t)(tBase + tt) * 64 + nt * 16,
             TLEN * 64, acc, lane);
}

extern "C" void kernel_launch(void* const* d_in, const int* in_sizes, int n_in,
                              void* d_out, int out_size, void* d_ws, size_t ws_size,
                              hipStream_t stream) {
  const float* state  = (const float*)d_in[0];
  const float* inputs = (const float*)d_in[1];
  const float* epsW   = (const float*)d_in[2];
  const float* epsV   = (const float*)d_in[3];
  const float* F      = (const float*)d_in[4];
  const float* Bmat   = (const float*)d_in[5];
  const float* H      = (const float*)d_in[6];
  const float* sqrtSW = (const float*)d_in[7];
  const float* sqrtSV = (const float*)d_in[8];

  float* states = (float*)d_out;                         // [64][2048][256]
  float* obs    = states + (size_t)BATCH * TLEN * S_DIM; // [64][2048][64]
  float* Pbase  = (float*)d_ws;                          // F^1..F^64 (16 MB)
  float* carry  = Pbase + (size_t)CHUNK * MATSZ;         // [31][64][256] (2 MB)

  copy_f_kernel<<<64, 256, 0, stream>>>(F, Pbase);       // P_1 = F
  for (int s = 1; s < CHUNK; s <<= 1)                    // log-depth doubling
    powers_kernel<<<s, 512, 256 * LDP * 4, stream>>>(Pbase, s);

  scan_kernel<<<NCHUNK * 4, 512, (256 * LDP + 48 * LDP + 16 * LDPI) * 4, stream>>>(
      state, inputs, epsW, F, Bmat, sqrtSW, states);

  carry_kernel<<<1, 512, 64 * LDP * 4, stream>>>(
      states, Pbase + (size_t)(CHUNK - 1) * MATSZ, carry);

  fixup_kernel<<<(NCHUNK - 1) * 4, 512, (256 * LDP + 16 * LDP) * 4, stream>>>(
      states, Pbase, carry);

  obs_kernel<<<(TLEN / 4) * 4, 512,
               (64 * LDP + 64 * LDPI + 64 * LDP + 64 * LDPI) * 4, stream>>>(
      states, epsV, H, sqrtSV, obs);
}